// MyModel_61933428410535
// MI455X (gfx1250) — compile-verified
//
#include <hip/hip_runtime.h>
#include <hip/hip_bf16.h>

// ifft(fft(x)) == identity; output is complex64 (re = x, im = 0).
// Pure bandwidth problem: 256 MiB in + 512 MiB out => ~33 us floor @ 23.3 TB/s.
// Optimal kernel = streaming expand-copy using the gfx1250 async LDS path
// (GLOBAL_LOAD_ASYNC_TO_LDS_B128, ASYNCcnt) with double buffering, and
// non-temporal 128-bit stores so we don't pollute L2 with write-once data.

typedef float float4v __attribute__((ext_vector_type(4)));

#define THREADS 256
#define GRID    8192

typedef __attribute__((address_space(3))) float4v* lds_f4_ptr;

__global__ __launch_bounds__(THREADS)
void fft_ifft_identity_kernel(const float4v* __restrict__ x4,
                              float4v* __restrict__ out4,
                              long long nvec4) {
    __shared__ float4v smem[2][THREADS];

    const int tid = threadIdx.x;
    const long long stride = (long long)GRID * THREADS;      // in float4 units
    long long idx = (long long)blockIdx.x * THREADS + tid;

    // Per-thread LDS stage slots as 32-bit LDS addresses (addrspace(3) VGPRs).
    lds_f4_ptr buf0 = (lds_f4_ptr)&smem[0][tid];
    lds_f4_ptr buf1 = (lds_f4_ptr)&smem[1][tid];

    // Prologue: kick off stage 0 (async copy global -> LDS, tracked by ASYNCcnt).
    if (idx < nvec4) {
        asm volatile("global_load_async_to_lds_b128 %0, %1, off"
                     :: "v"(buf0), "v"(x4 + idx) : "memory");
    }

    int cur = 0;
    while (idx < nvec4) {
        const long long nxt = idx + stride;
        const bool have_next = (nxt < nvec4);

        if (have_next) {
            lds_f4_ptr dst = cur ? buf0 : buf1;
            asm volatile("global_load_async_to_lds_b128 %0, %1, off"
                         :: "v"(dst), "v"(x4 + nxt) : "memory");
            // 2 async ops outstanding; wait until only the new one remains.
            asm volatile("s_wait_asynccnt 1" ::: "memory");
        } else {
            asm volatile("s_wait_asynccnt 0" ::: "memory");
        }

        // Read the completed stage from LDS (ds_load_b128).
        float4v v = smem[cur][tid];

        // Expand real -> interleaved complex: (a,b,c,d) -> (a,0,b,0),(c,0,d,0).
        float4v lo; lo.x = v.x; lo.y = 0.0f; lo.z = v.y; lo.w = 0.0f;
        float4v hi; hi.x = v.z; hi.y = 0.0f; hi.z = v.w; hi.w = 0.0f;

        // Non-temporal 128-bit stores: write-once stream, keep it out of L2 RT set.
        __builtin_nontemporal_store(lo, out4 + (idx << 1));
        __builtin_nontemporal_store(hi, out4 + (idx << 1) + 1);

        idx = nxt;
        cur ^= 1;
    }
}

extern "C" void kernel_launch(void* const* d_in, const int* in_sizes, int n_in,
                              void* d_out, int out_size, void* d_ws, size_t ws_size,
                              hipStream_t stream) {
    (void)n_in; (void)out_size; (void)d_ws; (void)ws_size;

    const float* x = (const float*)d_in[0];
    float* out = (float*)d_out;                 // complex64 interleaved (re, im)

    long long n = (long long)in_sizes[0];       // 16*1024*4096 = 67,108,864 floats
    long long nvec4 = n >> 2;                   // 16,777,216 float4 chunks

    fft_ifft_identity_kernel<<<GRID, THREADS, 0, stream>>>(
        (const float4v*)x, (float4v*)out, nvec4);
}